// GatingNetwork_34170759807254
// MI455X (gfx1250) — compile-verified
//
#include <hip/hip_runtime.h>

// ---------------------------------------------------------------------------
// MoE gating network for MI455X (gfx1250, wave32, WMMA)
//   logits = x @ W^T  (f32 WMMA 16x16x4, async-to-LDS double-buffered),
//   softmax over 64 experts, top-8 candidates -> top-2, 2-way softmax weights.
// ---------------------------------------------------------------------------

typedef __attribute__((ext_vector_type(2))) float v2f;
typedef __attribute__((ext_vector_type(8))) float v8f;

#define B_TOTAL   16384
#define D_DIM     2048
#define M_EXPERTS 64
#define KC        64      // K-chunk staged in LDS
#define NCHUNK    (D_DIM / KC)
#define TILE_ROWS 128     // rows per workgroup
#define THREADS   256     // 8 wave32 waves
#define LSTR      68      // padded LDS row stride (floats): conflict-free + 16B aligned

// Output byte offsets (return order: topk_idx i64, topk_w f32, probs f32, topc_idx i64)
#define TKI_BYTES ((size_t)B_TOTAL * 2 * 8)
#define TKW_BYTES ((size_t)B_TOTAL * 2 * 4)
#define PRB_BYTES ((size_t)B_TOTAL * 64 * 4)
#define TKW_OFF   (TKI_BYTES)
#define PRB_OFF   (TKI_BYTES + TKW_BYTES)
#define TCI_OFF   (TKI_BYTES + TKW_BYTES + PRB_BYTES)

// Async DMA: memory -> LDS, 16 bytes per lane, tracked with ASYNCcnt.
// LDS operand is the 32-bit LDS byte address (= low 32 bits of the flat
// shared-aperture pointer, per the flat-addressing LDS mapping rule).
__device__ __forceinline__ void async_copy_b128(const void* gaddr, void* lds_ptr) {
    unsigned l = (unsigned)(unsigned long long)lds_ptr;
    asm volatile("global_load_async_to_lds_b128 %0, %1, off"
                 :: "v"(l), "v"(gaddr) : "memory");
}

__device__ __forceinline__ void issue_tile_loads(const float* __restrict__ x,
                                                 const float* __restrict__ Wm,
                                                 float* xs_buf, float* ws_buf,
                                                 int tid, int rowBase, int kb) {
    // x tile: 128 rows x 64 cols = 2048 float4 segments, 8 per thread
    #pragma unroll
    for (int it = 0; it < 8; ++it) {
        int idx = tid + it * THREADS;
        int row = idx >> 4;
        int seg = idx & 15;
        async_copy_b128(x + (size_t)(rowBase + row) * D_DIM + kb + seg * 4,
                        &xs_buf[row * LSTR + seg * 4]);
    }
    // W tile: 64 rows x 64 cols = 1024 float4 segments, 4 per thread
    #pragma unroll
    for (int it = 0; it < 4; ++it) {
        int idx = tid + it * THREADS;
        int row = idx >> 4;
        int seg = idx & 15;
        async_copy_b128(Wm + (size_t)row * D_DIM + kb + seg * 4,
                        &ws_buf[row * LSTR + seg * 4]);
    }
}

__global__ __launch_bounds__(THREADS)
void gating_wmma_kernel(const float* __restrict__ x,
                        const float* __restrict__ Wm,
                        char* __restrict__ out) {
    // Double-buffered tiles (~102 KB of the 320 KB WGP LDS)
    __shared__ __align__(16) float xs[2][TILE_ROWS * LSTR];
    __shared__ __align__(16) float ws[2][M_EXPERTS * LSTR];

    const int tid     = threadIdx.x;
    const int wave    = tid >> 5;        // wave32
    const int lane    = tid & 31;
    const int half    = lane >> 4;       // 0: lanes 0-15, 1: lanes 16-31
    const int l16     = lane & 15;
    const int rowBase = blockIdx.x * TILE_ROWS;

    v8f acc[4];
    #pragma unroll
    for (int n = 0; n < 4; ++n) acc[n] = (v8f)0.0f;

    // ------------------------- GEMM main loop (async pipelined) ------------
    issue_tile_loads(x, Wm, xs[0], ws[0], tid, rowBase, 0);

    for (int c = 0; c < NCHUNK; ++c) {
        const int cur = c & 1;

        // this wave's chunk-c async copies have landed in LDS
        asm volatile("s_wait_asynccnt 0x0" ::: "memory");
        // all waves: chunk-c data visible, and everyone is done reading the
        // other buffer (their chunk c-1 compute precedes this barrier)
        __syncthreads();

        if (c + 1 < NCHUNK)
            issue_tile_loads(x, Wm, xs[cur ^ 1], ws[cur ^ 1],
                             tid, rowBase, (c + 1) * KC);

        // A fragment source: this wave's 16-row stripe
        const float* xrow = &xs[cur][(wave * 16 + l16) * LSTR + 2 * half];
        const float* wsb  = ws[cur];
        #pragma unroll
        for (int k = 0; k < KC; k += 4) {
            // A 16x4 f32: lane<16 -> K=k..k+1 ; lane>=16 -> K=k+2..k+3
            v2f a = *(const v2f*)(xrow + k);
            #pragma unroll
            for (int n = 0; n < 4; ++n) {
                // B 4x16: B[k][n] = W[n][k]; N striped over lanes, K split by half
                v2f b = *(const v2f*)(&wsb[(n * 16 + l16) * LSTR + 2 * half + k]);
                acc[n] = __builtin_amdgcn_wmma_f32_16x16x4_f32(
                    /*neg_a=*/false, a, /*neg_b=*/false, b,
                    /*c_mod=*/(short)0, acc[n],
                    /*reuse_a=*/false, /*reuse_b=*/false);
            }
        }
    }
    __syncthreads();

    // ----------------- dump logits to LDS (reuse xs[0]) --------------------
    // C/D layout: acc[n][i] = C[m = i + 8*half][n*16 + l16]
    float* ls = &xs[0][0];
    #pragma unroll
    for (int n = 0; n < 4; ++n) {
        #pragma unroll
        for (int i = 0; i < 8; ++i) {
            int m = i + 8 * half;
            ls[(wave * 16 + m) * LSTR + n * 16 + l16] = acc[n][i];
        }
    }
    // same-wave DS in-order; make the stores visible before cross-lane reads
    asm volatile("s_wait_dscnt 0x0" ::: "memory");

    // ----------------- per-row epilogue (lanes 0-15, 1 row each) -----------
    if (lane < 16) {
        const int rloc = wave * 16 + l16;
        const size_t row = (size_t)(rowBase + rloc);

        float v[64];
        #pragma unroll
        for (int j = 0; j < 16; ++j) {
            float4 t = *(const float4*)(&ls[rloc * LSTR + j * 4]);
            v[4 * j + 0] = t.x; v[4 * j + 1] = t.y;
            v[4 * j + 2] = t.z; v[4 * j + 3] = t.w;
        }

        // stable softmax over 64 experts (TAU == 1.0 -> logits unscaled)
        float maxv = v[0];
        #pragma unroll
        for (int j = 1; j < 64; ++j) maxv = fmaxf(maxv, v[j]);
        float esum = 0.0f;
        #pragma unroll
        for (int j = 0; j < 64; ++j) esum += __expf(v[j] - maxv);
        const float inv = 1.0f / (esum + 1e-12f);

        float* probs = (float*)(out + PRB_OFF);
        #pragma unroll
        for (int j = 0; j < 16; ++j) {
            float4 p;
            p.x = __expf(v[4 * j + 0] - maxv) * inv;
            p.y = __expf(v[4 * j + 1] - maxv) * inv;
            p.z = __expf(v[4 * j + 2] - maxv) * inv;
            p.w = __expf(v[4 * j + 3] - maxv) * inv;
            *(float4*)(probs + row * 64 + j * 4) = p;
        }

        // top-8 by repeated max; strictly-greater => lowest index wins ties
        float cv[8];
        int   ci[8];
        #pragma unroll
        for (int s = 0; s < 8; ++s) {
            float mv = -__builtin_inff();
            int   mi = 0;
            #pragma unroll
            for (int j = 0; j < 64; ++j) {
                if (v[j] > mv) { mv = v[j]; mi = j; }
            }
            cv[s] = mv; ci[s] = mi;
            #pragma unroll
            for (int j = 0; j < 64; ++j)
                v[j] = (j == mi) ? -__builtin_inff() : v[j];
        }

        // top-2 of the (already descending) top-8 are elements 0 and 1
        long long* tki = (long long*)out;
        float*     tkw = (float*)(out + TKW_OFF);
        long long* tci = (long long*)(out + TCI_OFF);

        tki[row * 2 + 0] = (long long)ci[0];
        tki[row * 2 + 1] = (long long)ci[1];

        // stable 2-way softmax: max = cv[0] (cv sorted descending)
        float e1 = __expf(cv[1] - cv[0]);
        float s2 = 1.0f + e1 + 1e-12f;
        tkw[row * 2 + 0] = 1.0f / s2;
        tkw[row * 2 + 1] = e1 / s2;

        #pragma unroll
        for (int s = 0; s < 8; ++s)
            tci[row * 8 + s] = (long long)ci[s];
    }
}

extern "C" void kernel_launch(void* const* d_in, const int* in_sizes, int n_in,
                              void* d_out, int out_size, void* d_ws, size_t ws_size,
                              hipStream_t stream) {
    const float* x = (const float*)d_in[0];   // [16384, 2048] f32
    const float* W = (const float*)d_in[1];   // [64, 2048]   f32
    dim3 grid(B_TOTAL / TILE_ROWS);           // 128 workgroups
    dim3 block(THREADS);                      // 8 waves (wave32)
    gating_wmma_kernel<<<grid, block, 0, stream>>>(x, W, (char*)d_out);
}